// CenterLoss_30588757082777
// MI455X (gfx1250) — compile-verified
//
#include <hip/hip_runtime.h>
#include <hip/hip_bf16.h>
#include <climits>

typedef __attribute__((ext_vector_type(2))) float v2f;
typedef __attribute__((ext_vector_type(8))) float v8f;

#define MU        0.5f
#define CLAMP_LO  1e-12f
#define CLAMP_HI  1e12f

// ---------------------------------------------------------------- kernel 0
// Re-initialize all scratch every call (harness poisons ws once, never again).
__global__ void cl_zero(float* __restrict__ sums, int* __restrict__ counts,
                        float* __restrict__ sdist, int* __restrict__ first,
                        int nSums, int C) {
    int i = blockIdx.x * blockDim.x + threadIdx.x;
    int stride = gridDim.x * blockDim.x;
    for (int j = i; j < nSums; j += stride) sums[j] = 0.0f;
    for (int j = i; j < C; j += stride) { counts[j] = 0; sdist[j] = 0.0f; }
    if (i == 0) *first = INT_MAX;
}

// ---------------------------------------------------------------- kernel 1
// segment_sum + counts. One block per row; fp32 atomics land in L2 (4 MB target).
__global__ void cl_accum(const float* __restrict__ x, const int* __restrict__ labels,
                         float* __restrict__ sums, int* __restrict__ counts, int D) {
    int b = blockIdx.x;
    int t = threadIdx.x;
    int l = labels[b];
    atomicAdd(&sums[(size_t)l * D + t], x[(size_t)b * D + t]);
    if (t == 0) atomicAdd(&counts[l], 1);
}

// ---------------------------------------------------------------- kernel 2
// first available class = min index with counts>0
__global__ void cl_first(const int* __restrict__ counts, int* __restrict__ first, int C) {
    int c = blockIdx.x * blockDim.x + threadIdx.x;
    if (c < C && counts[c] > 0) atomicMin(first, c);
}

// ---------------------------------------------------------------- kernel 3
// new_centers: first available class -> mean; other available -> EMA; else keep.
__global__ void cl_newcent(const float* __restrict__ centers, const float* __restrict__ sums,
                           const int* __restrict__ counts, const int* __restrict__ first,
                           float* __restrict__ nc, int C, int D) {
    int total = C * D;
    int f = *first;
    int stride = gridDim.x * blockDim.x;
    for (int i = blockIdx.x * blockDim.x + threadIdx.x; i < total; i += stride) {
        int c = i / D;
        int n = counts[c];
        float mean = sums[i] / (float)(n > 1 ? n : 1);
        float ctr  = centers[i];
        float v;
        if (n > 0) v = (c == f) ? mean : (1.0f - MU) * ctr + MU * mean;
        else       v = ctr;
        nc[i] = v;
    }
}

// ---------------------------------------------------------------- kernel 4
// One wave (32 lanes) per 16 rows. Gathered centers, dot products on the
// matrix pipe via V_WMMA_F32_16X16X4_F32; we consume the diagonal of the
// 16x16 cross-dot matrix. ||x||^2 / ||c||^2 accumulate on VALU alongside.
//
// fp32 A 16x4 layout: lanes 0-15 hold row M=lane  {K0 in vgpr0, K1 in vgpr1},
//                     lanes 16-31 hold row M=lane-16 {K2, K3}.
// fp32 B 4x16 layout mirrors it with N=lane&15. Both operands therefore use
// the identical per-lane addressing below (A from x rows, B from gathered nc).
__global__ void __launch_bounds__(128) cl_dist(const float* __restrict__ x,
                                               const float* __restrict__ nc,
                                               const int* __restrict__ labels,
                                               float* __restrict__ sdist,
                                               int B, int D) {
    __shared__ float cbuf[4][32][8];   // per-wave spill of the C/D accumulator

    int wave = threadIdx.x >> 5;
    int lane = threadIdx.x & 31;
    int group = blockIdx.x * 4 + wave;
    int nGroups = B >> 4;
    bool active = group < nGroups;

    int m    = lane & 15;        // row within group (and N column for B operand)
    int half = lane >> 4;        // 0 -> K{0,1}, 1 -> K{2,3} of each K=4 slice
    int lbl  = 0;
    float nxs = 0.0f, ncs = 0.0f;

    if (active) {
        int row = group * 16 + m;
        lbl = labels[row];
        const float* xr = x  + (size_t)row * D;
        const float* cr = nc + (size_t)lbl * D;
        int k0 = half * 2;

        v8f acc = {};
        float nx = 0.0f, nc2 = 0.0f;
        for (int kb = 0; kb < D; kb += 4) {
            float a0 = xr[kb + k0], a1 = xr[kb + k0 + 1];
            float b0 = cr[kb + k0], b1 = cr[kb + k0 + 1];
            nx  += a0 * a0 + a1 * a1;
            nc2 += b0 * b0 + b1 * b1;
            v2f A  = {a0, a1};
            v2f Bm = {b0, b1};
            acc = __builtin_amdgcn_wmma_f32_16x16x4_f32(
                false, A, false, Bm, (short)0, acc, false, false);
        }
#pragma unroll
        for (int j = 0; j < 8; ++j) cbuf[wave][lane][j] = acc[j];

        // full norms: lane m holds k={0,1} mod 4 partials, lane m+16 holds k={2,3}
        nxs = nx  + __shfl_down(nx,  16, 32);
        ncs = nc2 + __shfl_down(nc2, 16, 32);
    }
    __syncthreads();

    if (active && lane < 16) {
        // diagonal of C: m<8 -> vgpr m @ lane m; m>=8 -> vgpr m-8 @ lane m+16
        float dot = (m < 8) ? cbuf[wave][m][m] : cbuf[wave][m + 16][m - 8];
        float d2 = nxs + ncs - 2.0f * dot;
        d2 = fminf(fmaxf(d2, CLAMP_LO), CLAMP_HI);
        atomicAdd(&sdist[lbl], d2);
    }
}

// ---------------------------------------------------------------- kernel 5
// loss = mean_c[ (sum_dist[c] + (B-counts[c])*1e-12) / max(counts[c],1) ] / D
__global__ void cl_final(const float* __restrict__ sdist, const int* __restrict__ counts,
                         float* __restrict__ out, int B, int C, int D) {
    __shared__ float red[256];
    float acc = 0.0f;
    for (int c = threadIdx.x; c < C; c += blockDim.x) {
        int n = counts[c];
        float s = sdist[c] + (float)(B - n) * CLAMP_LO;
        acc += s / (float)(n > 1 ? n : 1);
    }
    red[threadIdx.x] = acc;
    __syncthreads();
    for (int s = 128; s > 0; s >>= 1) {
        if ((int)threadIdx.x < s) red[threadIdx.x] += red[threadIdx.x + s];
        __syncthreads();
    }
    if (threadIdx.x == 0) out[0] = red[0] / (float)C / (float)D;
}

// ----------------------------------------------------------------
extern "C" void kernel_launch(void* const* d_in, const int* in_sizes, int n_in,
                              void* d_out, int out_size, void* d_ws, size_t ws_size,
                              hipStream_t stream) {
    const float* x       = (const float*)d_in[0];
    const float* centers = (const float*)d_in[1];
    const int*   labels  = (const int*)d_in[2];

    int B = in_sizes[2];            // 16384
    int D = in_sizes[0] / B;        // 256
    int C = in_sizes[1] / D;        // 4096

    // workspace layout
    char* ws = (char*)d_ws;
    size_t cd = (size_t)C * D;
    float* sums   = (float*)ws;                       // C*D f32 (4 MB)
    float* ncent  = (float*)(ws + cd * 4);            // C*D f32 (4 MB)
    int*   counts = (int*)  (ws + 2 * cd * 4);        // C i32
    float* sdist  = (float*)(ws + 2 * cd * 4 + (size_t)C * 4);  // C f32
    int*   first  = (int*)  (ws + 2 * cd * 4 + (size_t)C * 8);  // 1 i32
    float* out    = (float*)d_out;

    cl_zero<<<1024, 256, 0, stream>>>(sums, counts, sdist, first, (int)cd, C);
    cl_accum<<<B, D, 0, stream>>>(x, labels, sums, counts, D);
    cl_first<<<(C + 255) / 256, 256, 0, stream>>>(counts, first, C);
    cl_newcent<<<1024, 256, 0, stream>>>(centers, sums, counts, first, ncent, C, D);
    cl_dist<<<(B / 16 + 3) / 4, 128, 0, stream>>>(x, ncent, labels, sdist, B, D);
    cl_final<<<1, 256, 0, stream>>>(sdist, counts, out, B, C, D);
}